// VariationalFFNFull_10857677324510
// MI455X (gfx1250) — compile-verified
//
#include <hip/hip_runtime.h>
#include <hip/hip_bf16.h>

// Problem constants fixed by the reference: B=8, H=8, N=512, K=16
constexpr int Bdim = 8;
constexpr int Hdim = 8;
constexpr int Ndim = 512;
constexpr int Kdim = 16;
constexpr float ALPHA_C = 0.001f;   // TAU = KAPPA = 1, N_ITER = 1

constexpr int CHUNK  = 128;  // j columns staged per iteration
constexpr int APITCH = 133;  // LDS pitch: conflict-free column reads (133 % 64 = 5, odd & coprime)

typedef __attribute__((ext_vector_type(2))) float v2f;
typedef __attribute__((ext_vector_type(8))) float v8f;

#if defined(__gfx1250__) && __has_builtin(__builtin_amdgcn_wmma_f32_16x16x4_f32)
#define USE_WMMA_F32 1
static __device__ inline v8f wmma_16x16x4(v2f a, v2f b, v8f c) {
  // 8 args: (neg_a, A, neg_b, B, c_mod, C, reuse_a, reuse_b)
  return __builtin_amdgcn_wmma_f32_16x16x4_f32(false, a, false, b, (short)0, c, false, false);
}
#else
#define USE_WMMA_F32 0
#endif

// ---------------------------------------------------------------------------
// Kernel A: per (b,n) 16x16 matrix exponential R = expm(sum_g phi_g G_g) via
// scaling-and-squaring + order-12 Taylor, plus v = R^T mu, |mu|^2, |v|^2.
// One 256-thread block per matrix (thread = one element). 4096 blocks total.
// ---------------------------------------------------------------------------
__global__ __launch_bounds__(256)
void expm_stage_kernel(const float* __restrict__ mu,
                       const float* __restrict__ phi,
                       const float* __restrict__ gen,
                       float* __restrict__ Rws,
                       float* __restrict__ Vws,
                       float* __restrict__ Aws,
                       float* __restrict__ Bws)
{
  const int n  = blockIdx.x;
  const int b  = blockIdx.y;
  const int tid = threadIdx.x;
  const int rr = tid >> 4;
  const int cc = tid & 15;
  const size_t bn = (size_t)b * Ndim + n;

  __shared__ float T[16][17];
  __shared__ float P[16][17];
  __shared__ float E[16][17];
  __shared__ float red[256];
  __shared__ float mus[16];
  __shared__ float vv[16];
  __shared__ float scal_s;
  __shared__ int   sexp_s;

  const float p0 = phi[bn*3+0], p1 = phi[bn*3+1], p2 = phi[bn*3+2];
  const float a  = p0 * gen[(0*16+rr)*16+cc]
                 + p1 * gen[(1*16+rr)*16+cc]
                 + p2 * gen[(2*16+rr)*16+cc];

  red[tid] = a * a;
  __syncthreads();
  for (int st = 128; st > 0; st >>= 1) {
    if (tid < st) red[tid] += red[tid + st];
    __syncthreads();
  }
  if (tid == 0) {
    float nrm = sqrtf(red[0]);
    int s = 0;
    while (nrm > 0.25f && s < 24) { nrm *= 0.5f; ++s; }
    sexp_s = s;
    scal_s = ldexpf(1.0f, -s);
  }
  if (tid < 16) mus[tid] = mu[bn*Kdim + tid];
  __syncthreads();

  T[rr][cc] = a * scal_s;
  const float id = (rr == cc) ? 1.0f : 0.0f;
  P[rr][cc] = id;
  E[rr][cc] = id;
  const int sexp = sexp_s;
  __syncthreads();

  // E = I + T + T^2/2! + ... + T^12/12!   (Horner-free running product)
  #pragma unroll 1
  for (int it = 1; it <= 12; ++it) {
    float acc = 0.0f;
    #pragma unroll
    for (int l = 0; l < 16; ++l) acc += P[rr][l] * T[l][cc];
    acc *= (1.0f / (float)it);
    __syncthreads();
    P[rr][cc] = acc;
    E[rr][cc] += acc;
    __syncthreads();
  }
  // undo scaling: E <- E^(2^s)
  #pragma unroll 1
  for (int q = 0; q < sexp; ++q) {
    float acc = 0.0f;
    #pragma unroll
    for (int l = 0; l < 16; ++l) acc += E[rr][l] * E[l][cc];
    __syncthreads();
    E[rr][cc] = acc;
    __syncthreads();
  }

  Rws[(bn*Kdim + rr)*Kdim + cc] = E[rr][cc];

  // v = R^T mu  (note expm(-A) == expm(A)^T for skew A, exactly incl. Pade)
  if (tid < 16) {
    float acc = 0.0f;
    #pragma unroll
    for (int l = 0; l < 16; ++l) acc += E[l][tid] * mus[l];
    vv[tid] = acc;
    Vws[bn*Kdim + tid] = acc;
  }
  __syncthreads();
  if (tid == 0) {
    float an = 0.0f, bn2 = 0.0f;
    #pragma unroll
    for (int k = 0; k < 16; ++k) { an += mus[k]*mus[k]; bn2 += vv[k]*vv[k]; }
    Aws[bn] = an;  // |mu_i|^2
    Bws[bn] = bn2; // |v_j|^2
  }
}

// ---------------------------------------------------------------------------
// Kernel C: stream beta (67 MB, the roofline term) exactly once.
// Block = (b, 16-row i-tile), 256 threads / 8 waves. Per 128-column chunk:
//   stage H-averaged beta tile + V chunk in LDS; each wave owns 16 columns,
//   computes kl/w in WMMA-A fragment layout, accumulates Sv += avg@V and
//   Wv += w@V with v_wmma_f32_16x16x4_f32 (persistent f32 accumulators).
// Then cross-wave LDS reduction + fused R_i rotation and final mu update.
// ---------------------------------------------------------------------------
__global__ __launch_bounds__(256)
void fused_update_kernel(const float* __restrict__ mu,
                         const float* __restrict__ beta,
                         const float* __restrict__ mu_prior,
                         const float* __restrict__ lr_p,
                         const float* __restrict__ Rws,
                         const float* __restrict__ Vws,
                         const float* __restrict__ Aws,
                         const float* __restrict__ Bws,
                         float* __restrict__ out)
{
  const int b    = blockIdx.y;
  const int i0   = blockIdx.x * 16;
  const int tid  = threadIdx.x;
  const int wave = tid >> 5;
  const int lane = tid & 31;
  const int row  = lane & 15;            // A-fragment: lane holds matrix row `row`
  const int off  = (lane < 16) ? 0 : 2;  // K-column offset within each 4-chunk

  __shared__ float avg_lds[16][APITCH];
  __shared__ float V_lds[CHUNK][Kdim];
  __shared__ float bn_lds[CHUNK];
  __shared__ float Sv_part[8][16][Kdim];
  __shared__ float Wv_part[8][16][Kdim];
  __shared__ float s_part[8][16];
  __shared__ float S0_part[8][16];
  __shared__ float Sv_red[16][Kdim];
  __shared__ float Wv_red[16][Kdim];
  __shared__ float s_red[16];
  __shared__ float S0_red[16];
#if !USE_WMMA_F32
  __shared__ float w_lds[16][APITCH];
#endif

  // per-lane: v row for this lane's i-row (for on-the-fly Gram dot products)
  float vi[Kdim];
  #pragma unroll
  for (int k = 0; k < Kdim; ++k)
    vi[k] = Vws[((size_t)b*Ndim + i0 + row)*Kdim + k];
  const float a_r = Aws[(size_t)b*Ndim + i0 + row];

  v8f sv = {0.f,0.f,0.f,0.f,0.f,0.f,0.f,0.f};
  v8f wv = {0.f,0.f,0.f,0.f,0.f,0.f,0.f,0.f};
  float s_acc = 0.0f, S0_acc = 0.0f;

  const float* beta_b = beta + (size_t)b * Hdim * Ndim * Ndim;

  for (int jt = 0; jt < Ndim / CHUNK; ++jt) {
    const int jbase = jt * CHUNK;
    __syncthreads();  // previous chunk's LDS reads are done

    // Coalesced stage of the H-averaged beta tile: 16 x 128, 8 elems/thread.
    for (int e = tid; e < 16*CHUNK; e += 256) {
      const int ii = e / CHUNK;
      const int jj = e % CHUNK;
      const float* bp = beta_b + (size_t)(i0 + ii)*Ndim + (jbase + jj);
      float acc = 0.0f;
      #pragma unroll
      for (int h = 0; h < Hdim; ++h) acc += bp[(size_t)h * Ndim * Ndim];
      avg_lds[ii][jj] = acc * (1.0f / (float)Hdim);
    }
    // Stage V chunk (128 x 16) and |v_j|^2
    for (int e = tid; e < CHUNK*Kdim; e += 256)
      V_lds[e >> 4][e & 15] = Vws[((size_t)b*Ndim + jbase)*Kdim + e];
    if (tid < CHUNK) bn_lds[tid] = Bws[(size_t)b*Ndim + jbase + tid];
    __syncthreads();

    // Each wave handles 16 columns: build A-fragments for beta_avg and w.
    float aw[8], ww[8];
    #pragma unroll
    for (int c = 0; c < 4; ++c) {
      #pragma unroll
      for (int u = 0; u < 2; ++u) {
        const int col = 16*wave + 4*c + off + u;
        const float av = avg_lds[row][col];
        float gm = 0.0f;                         // v_i . v_j (Gram, on the fly)
        #pragma unroll
        for (int k = 0; k < Kdim; ++k) gm += vi[k] * V_lds[col][k];
        const float kl   = 0.5f*(a_r + bn_lds[col]) - gm;
        const float wval = av * kl;
        aw[2*c+u] = av;
        ww[2*c+u] = wval;
        s_acc  += wval;   // row-partial of sum_j w
        S0_acc += av;     // row-partial of sum_j beta_avg
      }
    }

#if USE_WMMA_F32
    // Sv += avg @ V,  Wv += w @ V  (16x16x16 split into 4 k-chunks of 4).
    // EXEC is all-ones here (uniform control flow across the block).
    #pragma unroll
    for (int c = 0; c < 4; ++c) {
      const int colx = 16*wave + 4*c + off;
      v2f Am = { aw[2*c], aw[2*c+1] };
      v2f Wm = { ww[2*c], ww[2*c+1] };
      v2f Bm = { V_lds[colx][row], V_lds[colx+1][row] };  // B[K,N]: N = row
      sv = wmma_16x16x4(Am, Bm, sv);
      wv = wmma_16x16x4(Wm, Bm, wv);
    }
#else
    // Scalar fallback with identical C-layout semantics.
    #pragma unroll
    for (int c = 0; c < 4; ++c)
      #pragma unroll
      for (int u = 0; u < 2; ++u)
        w_lds[row][16*wave + 4*c + off + u] = ww[2*c+u];
    __syncthreads();
    #pragma unroll
    for (int e = 0; e < 8; ++e) {
      const int irow = e + ((lane < 16) ? 0 : 8);
      float acc1 = sv[e], acc2 = wv[e];
      #pragma unroll
      for (int jj = 0; jj < 16; ++jj) {
        const float bvv = V_lds[16*wave + jj][row];
        acc1 += avg_lds[irow][16*wave + jj] * bvv;
        acc2 += w_lds[irow][16*wave + jj]   * bvv;
      }
      sv[e] = acc1; wv[e] = acc2;
    }
#endif
  }

  // Complete per-row sums: row r lives in lanes r and r+16.
  s_acc  += __shfl_xor(s_acc, 16, 32);
  S0_acc += __shfl_xor(S0_acc, 16, 32);
  if (lane < 16) { s_part[wave][row] = s_acc; S0_part[wave][row] = S0_acc; }

  // Spill WMMA C-layout accumulators: VGPR e -> (M = e + 8*(lane>=16), N = row)
  #pragma unroll
  for (int e = 0; e < 8; ++e) {
    const int irow = e + ((lane < 16) ? 0 : 8);
    Sv_part[wave][irow][row] = sv[e];
    Wv_part[wave][irow][row] = wv[e];
  }
  __syncthreads();

  {
    const int i = tid >> 4, k = tid & 15;
    float r1 = 0.0f, r2 = 0.0f;
    #pragma unroll
    for (int w2 = 0; w2 < 8; ++w2) { r1 += Sv_part[w2][i][k]; r2 += Wv_part[w2][i][k]; }
    Sv_red[i][k] = r1; Wv_red[i][k] = r2;
    if (tid < 16) {
      float t1 = 0.0f, t2 = 0.0f;
      #pragma unroll
      for (int w2 = 0; w2 < 8; ++w2) { t1 += s_part[w2][tid]; t2 += S0_part[w2][tid]; }
      s_red[tid] = t1; S0_red[tid] = t2;
    }
  }
  __syncthreads();

  // Final fused update: rotate by R_i, assemble grads, write out.
  {
    const int i = tid >> 4, k = tid & 15;
    const int gi = i0 + i;
    const float* Rrow = Rws + (((size_t)b*Ndim + gi)*Kdim + k)*Kdim;  // R[b,gi][k][:]
    float rs = 0.0f, rw = 0.0f;
    #pragma unroll
    for (int l = 0; l < Kdim; ++l) { rs += Rrow[l]*Sv_red[i][l]; rw += Rrow[l]*Wv_red[i][l]; }
    const size_t idx = ((size_t)b*Ndim + gi)*Kdim + k;
    const float u  = mu[idx];
    const float up = mu_prior[idx];
    const float g1 = S0_red[i]*u - rs;        // grad1 (TAU=1); also bsum
    const float t1 = s_red[i]*u - rw;         // term1
    const float g2 = t1 - s_red[i]*g1;        // grad2 (KAPPA=1)
    const float lr = lr_p[0];
    out[idx] = u - lr * (ALPHA_C*(u - up) + g1 + g2);
  }
}

// ---------------------------------------------------------------------------
extern "C" void kernel_launch(void* const* d_in, const int* in_sizes, int n_in,
                              void* d_out, int out_size, void* d_ws, size_t ws_size,
                              hipStream_t stream) {
  (void)in_sizes; (void)n_in; (void)out_size; (void)ws_size;
  const float* mu       = (const float*)d_in[0];
  const float* beta     = (const float*)d_in[1];
  const float* mu_prior = (const float*)d_in[2];
  const float* phi      = (const float*)d_in[3];
  const float* gen      = (const float*)d_in[4];
  const float* lr       = (const float*)d_in[5];
  float* out = (float*)d_out;

  // Workspace layout (floats): R (B*N*K*K = 4 MB) | v (B*N*K) | |mu|^2 | |v|^2
  float* Rws = (float*)d_ws;
  float* Vws = Rws + (size_t)Bdim*Ndim*Kdim*Kdim;
  float* Aws = Vws + (size_t)Bdim*Ndim*Kdim;
  float* Bws = Aws + (size_t)Bdim*Ndim;

  expm_stage_kernel<<<dim3(Ndim, Bdim), dim3(256), 0, stream>>>(
      mu, phi, gen, Rws, Vws, Aws, Bws);
  fused_update_kernel<<<dim3(Ndim/16, Bdim), dim3(256), 0, stream>>>(
      mu, beta, mu_prior, lr, Rws, Vws, Aws, Bws, out);
}